// FeatureQuantizer_78108275245194
// MI455X (gfx1250) — compile-verified
//
#include <hip/hip_runtime.h>
#include <hip/hip_bf16.h>
#include <math.h>

typedef float v2f __attribute__((ext_vector_type(2)));
typedef float v8f __attribute__((ext_vector_type(8)));

#define D       256
#define KSTEPS  64          // D / 4 (K=4 per fp32 WMMA)
#define BSTRIDE 260         // LDS row stride (floats), conflict-free for frag loads

// ---- workspace layout (float offsets) ----
#define WS_HIST_C 0         // 1024
#define WS_HIST_F 1024      // 8192
#define WS_NORM_C 9216      // 1024
#define WS_NORM_F 10240     // 8192
#define WS_IDX_C  18432     // 192 ints
#define WS_IDX_F  18624     // 49152 ints
#define WS_SSE_C  67776     // 192
#define WS_SSE_F  67968     // 49152

// ---- output layout (float offsets) ----
#define Q_SIZE    (257 * 64 * 3 * 256)
#define OUT_LOSS  0
#define OUT_Q     1
#define OUT_FPERP (1 + Q_SIZE)
#define OUT_CPERP (2 + Q_SIZE)
#define OUT_IDX   (3 + Q_SIZE)

#define LEX_LT(sa, ia, sb, ib) (((sa) < (sb)) || (((sa) == (sb)) && ((ia) < (ib))))

#define INSERT_LEX(s, i, S0, I0, S1, I1, S2, I2)                 \
  do {                                                           \
    if (LEX_LT(s, i, S0, I0)) {                                  \
      S2 = S1; I2 = I1; S1 = S0; I1 = I0; S0 = (s); I0 = (i);    \
    } else if (LEX_LT(s, i, S1, I1)) {                           \
      S2 = S1; I2 = I1; S1 = (s); I1 = (i);                      \
    } else if (LEX_LT(s, i, S2, I2)) {                           \
      S2 = (s); I2 = (i);                                        \
    }                                                            \
  } while (0)

__device__ __forceinline__ float wave_sum32(float v) {
  #pragma unroll
  for (int o = 16; o > 0; o >>= 1) v += __shfl_xor(v, o, 32);
  return v;
}

// CDNA5 async memory->LDS staging: 64 bytes per lane (4 x b128), ASYNCcnt-tracked.
__device__ __forceinline__ void stage_async64(uint32_t lds_byte, uint32_t voff,
                                              const float* base) {
  asm volatile(
      "global_load_async_to_lds_b128 %0, %1, %2\n\t"
      "global_load_async_to_lds_b128 %0, %1, %2 offset:16\n\t"
      "global_load_async_to_lds_b128 %0, %1, %2 offset:32\n\t"
      "global_load_async_to_lds_b128 %0, %1, %2 offset:48"
      :
      : "v"(lds_byte), "v"(voff), "s"(base)
      : "memory");
}

__device__ __forceinline__ void wait_async0() {
#if __has_builtin(__builtin_amdgcn_s_wait_asynccnt)
  __builtin_amdgcn_s_wait_asynccnt(0);
#else
  asm volatile("s_wait_asynccnt 0x0" ::: "memory");
#endif
}

// Zero a float region of ws (histograms)
__global__ void init_kernel(float* ws, int n) {
  for (int i = blockIdx.x * blockDim.x + threadIdx.x; i < n;
       i += gridDim.x * blockDim.x)
    ws[i] = 0.0f;
}

// One wave per embedding row: ||e||^2
__global__ void norms_kernel(const float* __restrict__ emb,
                             float* __restrict__ norm, int E) {
  const int lane = threadIdx.x & 31;
  const int row  = blockIdx.x * 8 + (threadIdx.x >> 5);
  if (row >= E) return;
  const float* p = emb + (long)row * D;
  float s = 0.0f;
  #pragma unroll
  for (int j = 0; j < 8; ++j) {
    float v = p[lane + 32 * j];
    s += v * v;
  }
  s = wave_sum32(s);
  if (lane == 0) norm[row] = s;
}

// Fused distance-GEMM (fp32 WMMA) + streaming top-3 per row.
// Embedding tiles staged via async global->LDS, double-buffered.
__global__ __launch_bounds__(256) void vq_topk_kernel(
    const float* __restrict__ inp,    // [nRows][256]
    const float* __restrict__ emb,    // [E][256]
    const float* __restrict__ enorm,  // [E]
    int nRows, int E,
    int*   __restrict__ topIdx,       // [nRows][3]
    float* __restrict__ hist,         // [E]
    float* __restrict__ outIdx,       // d_out indices base for this segment
    int idxShift) {
  __shared__ float bTile[2][16 * BSTRIDE];

  const int tid   = threadIdx.x;
  const int lane  = tid & 31;
  const int wave  = tid >> 5;
  const int m     = lane & 15;        // row (A) / col (B,C) within tile
  const int khalf = (lane >> 4) * 2;  // K sub-offset for upper half-wave
  const int rowBase = (blockIdx.x * 8 + wave) * 16;
  const bool validWave = (rowBase < nRows);

  // Per-thread staging coordinates (16 codes x 256 floats per tile; 64B/thread)
  const int erow = tid >> 4;
  const int colB = (tid & 15) << 4;
  const uint32_t gByteBase = (uint32_t)((erow * D + colB) * 4);
  const uint32_t ldsByte0 =
      (uint32_t)(uintptr_t)(&bTile[0][erow * BSTRIDE + colB]);
  const uint32_t ldsByte1 =
      (uint32_t)(uintptr_t)(&bTile[1][erow * BSTRIDE + colB]);

  // A fragments: this wave's 16x256 row tile, resident in VGPRs.
  int rowG = rowBase + m;
  if (rowG >= nRows) rowG = nRows - 1;
  const float* arow = inp + (long)rowG * D;
  v2f areg[KSTEPS];
  #pragma unroll
  for (int kk = 0; kk < KSTEPS; ++kk)
    areg[kk] = *(const v2f*)(arow + 4 * kk + khalf);

  // Per-lane top-3 (8 independent rows per lane: accumulator VGPR r -> row)
  float ts0[8], ts1[8], ts2[8];
  int   ti0[8], ti1[8], ti2[8];
  #pragma unroll
  for (int r = 0; r < 8; ++r) {
    ts0[r] = ts1[r] = ts2[r] = 3.4e38f;
    ti0[r] = ti1[r] = ti2[r] = 0x7fffffff;
  }

  const int nTiles = E >> 4;

  // Prologue: stage tile 0 into buffer 0
  stage_async64(ldsByte0, gByteBase, emb);
  wait_async0();
  __syncthreads();

  for (int t = 0; t < nTiles; ++t) {
    const int e0 = t << 4;
    // Stage next tile into the other buffer while we compute on this one.
    if (t + 1 < nTiles) {
      const uint32_t voff = (uint32_t)((t + 1) << 14) + gByteBase;  // +16*256*4
      stage_async64(((t + 1) & 1) ? ldsByte1 : ldsByte0, voff, emb);
    }

    v8f acc = {0.f, 0.f, 0.f, 0.f, 0.f, 0.f, 0.f, 0.f};
    const float* bbase = &bTile[t & 1][m * BSTRIDE + khalf];
    #pragma unroll
    for (int kk = 0; kk < KSTEPS; ++kk) {
      v2f b = *(const v2f*)(bbase + 4 * kk);
      acc = __builtin_amdgcn_wmma_f32_16x16x4_f32(
          false, areg[kk], false, b, (short)0, acc, false, false);
    }

    // score = ||e||^2 - 2 x.e  (||x||^2 is row-constant, irrelevant to rank)
    const int eIdx = e0 + m;
    const float en = enorm[eIdx];
    #pragma unroll
    for (int r = 0; r < 8; ++r) {
      const float s = en - 2.0f * acc[r];
      // strict < keeps earliest index on ties (scan is in increasing index)
      if (s < ts0[r]) {
        ts2[r] = ts1[r]; ti2[r] = ti1[r];
        ts1[r] = ts0[r]; ti1[r] = ti0[r];
        ts0[r] = s;      ti0[r] = eIdx;
      } else if (s < ts1[r]) {
        ts2[r] = ts1[r]; ti2[r] = ti1[r];
        ts1[r] = s;      ti1[r] = eIdx;
      } else if (s < ts2[r]) {
        ts2[r] = s;      ti2[r] = eIdx;
      }
    }

    // My async writes for t+1 are done; barrier orders:
    //  (a) everyone's reads of bTile[t&1] complete before its overwrite at t+1
    //  (b) everyone's async writes of bTile[(t+1)&1] visible before reads at t+1
    wait_async0();
    __syncthreads();
  }

  // Butterfly merge of top-3 across the 16 lanes sharing each row
  // (masks 1,2,4,8 stay within each half-wave; rows = r (+8 for high half))
  #pragma unroll
  for (int r = 0; r < 8; ++r) {
    #pragma unroll
    for (int mask = 1; mask <= 8; mask <<= 1) {
      float os0 = __shfl_xor(ts0[r], mask, 32);
      int   oi0 = __shfl_xor(ti0[r], mask, 32);
      float os1 = __shfl_xor(ts1[r], mask, 32);
      int   oi1 = __shfl_xor(ti1[r], mask, 32);
      float os2 = __shfl_xor(ts2[r], mask, 32);
      int   oi2 = __shfl_xor(ti2[r], mask, 32);
      INSERT_LEX(os0, oi0, ts0[r], ti0[r], ts1[r], ti1[r], ts2[r], ti2[r]);
      INSERT_LEX(os1, oi1, ts0[r], ti0[r], ts1[r], ti1[r], ts2[r], ti2[r]);
      INSERT_LEX(os2, oi2, ts0[r], ti0[r], ts1[r], ti1[r], ts2[r], ti2[r]);
    }
  }

  // Lanes 0 and 16 publish the 8 rows of their half
  if (validWave && m == 0) {
    const int half = lane >> 4;
    #pragma unroll
    for (int r = 0; r < 8; ++r) {
      const int row = rowBase + r + 8 * half;
      if (row < nRows) {
        const int base = row * 3;
        topIdx[base + 0] = ti0[r];
        topIdx[base + 1] = ti1[r];
        topIdx[base + 2] = ti2[r];
        outIdx[base + 0] = (float)(ti0[r] + idxShift);
        outIdx[base + 1] = (float)(ti1[r] + idxShift);
        outIdx[base + 2] = (float)(ti2[r] + idxShift);
        atomicAdd(&hist[ti0[r]], 1.0f);
        atomicAdd(&hist[ti1[r]], 1.0f);
        atomicAdd(&hist[ti2[r]], 1.0f);
      }
    }
  }
}

// One wave per (row, k): quantized output gather + per-pair SSE
__global__ void gather_kernel(const float* __restrict__ inp,
                              const float* __restrict__ emb,
                              const int* __restrict__ topIdx,
                              float* __restrict__ outQ,
                              float* __restrict__ ssePair, int nPairs) {
  const int lane = threadIdx.x & 31;
  const int pair = blockIdx.x * 8 + (threadIdx.x >> 5);
  if (pair >= nPairs) return;
  const int  row = pair / 3;
  const long idx = topIdx[pair];
  const float* e = emb + idx * D;
  const float* x = inp + (long)row * D;
  float sse = 0.0f;
  #pragma unroll
  for (int j = 0; j < 8; ++j) {
    const int c = lane + 32 * j;
    const float v = e[c];
    const float d = v - x[c];
    sse += d * d;
    outQ[(long)pair * D + c] = v;
  }
  sse = wave_sum32(sse);
  if (lane == 0) ssePair[pair] = sse;
}

// Deterministic reductions: loss + both perplexities
__global__ void finalize_kernel(const float* __restrict__ ws,
                                float* __restrict__ out) {
  __shared__ float red[256];
  const int tid = threadIdx.x;
  float a;

  // feature perplexity: p = count / 16384
  a = 0.0f;
  for (int i = tid; i < 8192; i += 256) {
    float p = ws[WS_HIST_F + i] * (1.0f / 16384.0f);
    a += p * logf(p + 1e-10f);
  }
  red[tid] = a; __syncthreads();
  for (int s = 128; s > 0; s >>= 1) { if (tid < s) red[tid] += red[tid + s]; __syncthreads(); }
  const float fper = expf(-red[0]); __syncthreads();

  // class perplexity: p = count / 64
  a = 0.0f;
  for (int i = tid; i < 1024; i += 256) {
    float p = ws[WS_HIST_C + i] * (1.0f / 64.0f);
    a += p * logf(p + 1e-10f);
  }
  red[tid] = a; __syncthreads();
  for (int s = 128; s > 0; s >>= 1) { if (tid < s) red[tid] += red[tid + s]; __syncthreads(); }
  const float cper = expf(-red[0]); __syncthreads();

  // SSE sums
  a = 0.0f;
  for (int i = tid; i < 192; i += 256) a += ws[WS_SSE_C + i];
  red[tid] = a; __syncthreads();
  for (int s = 128; s > 0; s >>= 1) { if (tid < s) red[tid] += red[tid + s]; __syncthreads(); }
  const float sseC = red[0]; __syncthreads();

  a = 0.0f;
  for (int i = tid; i < 49152; i += 256) a += ws[WS_SSE_F + i];
  red[tid] = a; __syncthreads();
  for (int s = 128; s > 0; s >>= 1) { if (tid < s) red[tid] += red[tid + s]; __syncthreads(); }
  const float sseF = red[0];

  if (tid == 0) {
    // forward value: q_latent == e_latent -> loss = 1.25 * mean per codebook
    out[OUT_LOSS]  = 1.25f * (sseC / 49152.0f + sseF / 12582912.0f);
    out[OUT_FPERP] = fper;
    out[OUT_CPERP] = cper;
  }
}

extern "C" void kernel_launch(void* const* d_in, const int* in_sizes, int n_in,
                              void* d_out, int out_size, void* d_ws, size_t ws_size,
                              hipStream_t stream) {
  const float* features = (const float*)d_in[0];  // [257,64,256]
  const float* cemb     = (const float*)d_in[1];  // [1024,256]
  const float* femb     = (const float*)d_in[2];  // [8192,256]
  float* out = (float*)d_out;
  float* wsf = (float*)d_ws;
  int*   wsi = (int*)d_ws;

  // zero histograms
  init_kernel<<<36, 256, 0, stream>>>(wsf, 9216);

  // code norms
  norms_kernel<<<128, 256, 0, stream>>>(cemb, wsf + WS_NORM_C, 1024);
  norms_kernel<<<1024, 256, 0, stream>>>(femb, wsf + WS_NORM_F, 8192);

  // VQ top-3 (class: 64 rows x 1024 codes; feature: 16384 rows x 8192 codes)
  vq_topk_kernel<<<1, 256, 0, stream>>>(
      features, cemb, wsf + WS_NORM_C, 64, 1024,
      wsi + WS_IDX_C, wsf + WS_HIST_C, out + OUT_IDX, 0);
  vq_topk_kernel<<<128, 256, 0, stream>>>(
      features + 64 * 256, femb, wsf + WS_NORM_F, 16384, 8192,
      wsi + WS_IDX_F, wsf + WS_HIST_F, out + OUT_IDX + 64 * 3, 1024);

  // quantized gather + per-pair SSE
  gather_kernel<<<24, 256, 0, stream>>>(
      features, cemb, wsi + WS_IDX_C, out + OUT_Q, wsf + WS_SSE_C, 192);
  gather_kernel<<<6144, 256, 0, stream>>>(
      features + 64 * 256, femb, wsi + WS_IDX_F,
      out + OUT_Q + 64 * 3 * 256, wsf + WS_SSE_F, 49152);

  // scalars
  finalize_kernel<<<1, 256, 0, stream>>>(wsf, out);
}